// MyHeteroGNN_26010321944878
// MI455X (gfx1250) — compile-verified
//
#include <hip/hip_runtime.h>

#define NU 50000
#define NI 50000
#define D  128

typedef __attribute__((ext_vector_type(2))) float v2f;
typedef __attribute__((ext_vector_type(8))) float v8f;

__device__ __forceinline__ v8f wmma4(v2f a, v2f b, v8f c) {
  // D = A(16x4,f32) * B(4x16,f32) + C(16x16,f32)
  return __builtin_amdgcn_wmma_f32_16x16x4_f32(
      /*neg_a=*/false, a, /*neg_b=*/false, b,
      /*c_mod=*/(short)0, c, /*reuse_a=*/false, /*reuse_b=*/false);
}

// ---------------------------------------------------------------------------
// Workspace zeroing (agg + cnt accumulators must be re-zeroed every call).
// ---------------------------------------------------------------------------
__global__ __launch_bounds__(256) void zero_ws_kernel(float4* __restrict__ p, long n4) {
  long i = (long)blockIdx.x * blockDim.x + threadIdx.x;
  long stride = (long)gridDim.x * blockDim.x;
  float4 z = make_float4(0.f, 0.f, 0.f, 0.f);
  for (; i < n4; i += stride) p[i] = z;
}

// ---------------------------------------------------------------------------
// Edge scatter: one wave32 per edge; each lane carries 4 floats (float4 load,
// 4x global_atomic_add_f32). Aggregation buffers are L2-resident (192MB L2).
// ---------------------------------------------------------------------------
__global__ __launch_bounds__(256) void edge_scatter_kernel(
    const float* __restrict__ xsrc, const int* __restrict__ ei, int E,
    float* __restrict__ agg, float* __restrict__ cnt) {
  int gw = (int)(((unsigned)blockIdx.x * blockDim.x + threadIdx.x) >> 5);
  int lane = threadIdx.x & 31;
  if (gw >= E) return;
  int src = ei[gw];        // row 0 of [2,E]
  int dst = ei[E + gw];    // row 1 of [2,E]
  float4 v = ((const float4*)(xsrc + (size_t)src * D))[lane];
  float* a = agg + (size_t)dst * D + lane * 4;
  unsafeAtomicAdd(a + 0, v.x);
  unsafeAtomicAdd(a + 1, v.y);
  unsafeAtomicAdd(a + 2, v.z);
  unsafeAtomicAdd(a + 3, v.w);
  if (lane == 0) unsafeAtomicAdd(cnt + dst, 1.0f);
}

// ---------------------------------------------------------------------------
// Fused SAGE combine GEMM:
//   NS==2 (item): out = mean0 @ W0^T + A1 @ W1^T + bias0
//   NS==3 (user): out = 0.5*( mean0@W0^T + mean1@W1^T + A2@(W2a+W2b)^T
//                             + bias0 + bias1 )
// Block = 256 threads = 8 waves; block owns 16 output rows; wave w owns output
// cols [16w,16w+16). K=128 swept in steps of 4 with v_wmma_f32_16x16x4_f32.
// A panels staged in LDS (rows padded to 132 floats -> conflict-free stride-4
// bank pattern); mean division fused into the stage-in.
// ---------------------------------------------------------------------------
template <int NS>
__global__ __launch_bounds__(256) void sage_gemm_kernel(
    const float* __restrict__ A0, const float* __restrict__ c0, const float* __restrict__ W0,
    const float* __restrict__ A1, const float* __restrict__ c1, const float* __restrict__ W1,
    const float* __restrict__ A2, const float* __restrict__ W2a, const float* __restrict__ W2b,
    const float* __restrict__ bias0, const float* __restrict__ bias1,
    float scale, float* __restrict__ out) {
  __shared__ float As[NS][16][132];

  const int tid = threadIdx.x;
  const int m0 = blockIdx.x * 16;

  // Stage A panels: per stream 16x128 floats = 512 float4, 256 threads -> 2 each.
  for (int s = 0; s < NS; ++s) {
    const float* src = (s == 0) ? A0 : (s == 1) ? A1 : A2;
    const float* cnt = (s == 0) ? c0 : (s == 1) ? c1 : nullptr;
    for (int i = tid; i < 512; i += 256) {
      int r = i >> 5;        // row 0..15
      int c4 = i & 31;       // float4 col 0..31
      int row = m0 + r;
      float4 v = ((const float4*)(src + (size_t)row * D))[c4];
      if (cnt) {
        float inv = 1.0f / fmaxf(cnt[row], 1.0f);
        v.x *= inv; v.y *= inv; v.z *= inv; v.w *= inv;
      }
      *(float4*)&As[s][r][c4 * 4] = v;
    }
  }
  __syncthreads();

  const int wave = tid >> 5;
  const int lane = tid & 31;
  const int n0 = wave * 16;
  const int n = lane & 15;            // N within tile (B/C/D lane mapping)
  const int mr = lane & 15;           // M row for A fragment
  const int khalf = (lane >> 4) * 2;  // upper half-wave carries K+2..K+3
  const int ncol = n0 + n;            // global output column

  v8f acc = {};

#pragma unroll
  for (int kk = 0; kk < 128; kk += 4) {
    const int kb = kk + khalf;
    // stream 0
    {
      v2f a = *(const v2f*)&As[0][mr][kb];
      float2 w = *(const float2*)(W0 + (size_t)ncol * D + kb);  // B[k][n] = W[n][k]
      v2f b = {w.x, w.y};
      acc = wmma4(a, b, acc);
    }
    // stream 1
    {
      v2f a = *(const v2f*)&As[1][mr][kb];
      float2 w = *(const float2*)(W1 + (size_t)ncol * D + kb);
      v2f b = {w.x, w.y};
      acc = wmma4(a, b, acc);
    }
    // stream 2: self term vs summed weights (x @ (W2a+W2b)^T)
    if (NS == 3) {
      v2f a = *(const v2f*)&As[2][mr][kb];
      float2 wa = *(const float2*)(W2a + (size_t)ncol * D + kb);
      float2 wb = *(const float2*)(W2b + (size_t)ncol * D + kb);
      v2f b = {wa.x + wb.x, wa.y + wb.y};
      acc = wmma4(a, b, acc);
    }
  }

  float bval = bias0[ncol];
  if (NS == 3) bval += bias1[ncol];

  const int mhalf = (lane >> 4) * 8;  // C/D: VGPR r -> M = r + 8*(lane>=16)
  size_t base = (size_t)(m0 + mhalf) * D + ncol;
#pragma unroll
  for (int r = 0; r < 8; ++r)
    out[base + (size_t)r * D] = scale * (acc[r] + bval);
}

// ---------------------------------------------------------------------------
extern "C" void kernel_launch(void* const* d_in, const int* in_sizes, int n_in,
                              void* d_out, int out_size, void* d_ws, size_t ws_size,
                              hipStream_t stream) {
  const float* x_user = (const float*)d_in[0];
  const float* x_item = (const float*)d_in[1];
  const int* ei_rates = (const int*)d_in[2];
  const int* ei_rev   = (const int*)d_in[3];
  const int* ei_fol   = (const int*)d_in[4];
  const float* Wl_ra = (const float*)d_in[5];
  const float* bl_ra = (const float*)d_in[6];
  const float* Wr_ra = (const float*)d_in[7];
  const float* Wl_re = (const float*)d_in[8];
  const float* bl_re = (const float*)d_in[9];
  const float* Wr_re = (const float*)d_in[10];
  const float* Wl_fo = (const float*)d_in[11];
  const float* bl_fo = (const float*)d_in[12];
  const float* Wr_fo = (const float*)d_in[13];

  const int E = in_sizes[2] / 2;

  float* out_user = (float*)d_out;
  float* out_item = out_user + (size_t)NU * D;

  // workspace layout
  float* aggRa = (float*)d_ws;              // [NI, D]
  float* aggRe = aggRa + (size_t)NI * D;    // [NU, D]
  float* aggFo = aggRe + (size_t)NU * D;    // [NU, D]
  float* cntRa = aggFo + (size_t)NU * D;    // [NI]
  float* cntRe = cntRa + NI;                // [NU]
  float* cntFo = cntRe + NU;                // [NU]

  const long ztot = (long)NI * D + 2L * NU * D + NI + 2L * NU;  // floats (mult of 4)
  const long n4 = ztot / 4;
  int zb = (int)((n4 + 255) / 256);
  zero_ws_kernel<<<zb, 256, 0, stream>>>((float4*)d_ws, n4);

  // scatter: 8 edges per 256-thread block (wave per edge)
  int sb = (E + 7) / 8;
  edge_scatter_kernel<<<sb, 256, 0, stream>>>(x_user, ei_rates, E, aggRa, cntRa);
  edge_scatter_kernel<<<sb, 256, 0, stream>>>(x_item, ei_rev,   E, aggRe, cntRe);
  edge_scatter_kernel<<<sb, 256, 0, stream>>>(x_user, ei_fol,   E, aggFo, cntFo);

  // combine GEMMs: 50000/16 = 3125 row-tiles
  dim3 gg(NU / 16);
  sage_gemm_kernel<2><<<gg, 256, 0, stream>>>(
      aggRa, cntRa, Wl_ra, x_item, nullptr, Wr_ra,
      nullptr, nullptr, nullptr, bl_ra, nullptr, 1.0f, out_item);
  sage_gemm_kernel<3><<<gg, 256, 0, stream>>>(
      aggRe, cntRe, Wl_re, aggFo, cntFo, Wl_fo,
      x_user, Wr_re, Wr_fo, bl_re, bl_fo, 0.5f, out_user);
}